// GIN_78331613544734
// MI455X (gfx1250) — compile-verified
//
#include <hip/hip_runtime.h>
#include <hip/hip_bf16.h>

#define N_NODES  100000
#define N_EDGES  1600000
#define N_GRAPHS 1024
#define IN_DIM   79
#define IN_PAD   80
#define HIDDEN   128
#define OUT_DIM  50
#define OUT_PAD  64
#define BN_EPS   1e-5f

typedef __attribute__((ext_vector_type(2))) float v2f;
typedef __attribute__((ext_vector_type(8))) float v8f;

// ---------------------------------------------------------------- utilities
__global__ void zero_f32(float* __restrict__ p, long long n) {
    long long i = (long long)blockIdx.x * blockDim.x + threadIdx.x;
    if (i < n) p[i] = 0.f;
}

// dst (row-major, same width) = src for first n_src elems, 0 for the rest
__global__ void copy_zero_tail(const float* __restrict__ src, float* __restrict__ dstp,
                               int n_src, int n_total) {
    int i = blockIdx.x * blockDim.x + threadIdx.x;
    if (i >= n_total) return;
    dstp[i] = (i < n_src) ? src[i] : 0.f;
}

// dst[r*NP + c] = (c < N0) ? src[r*N0 + c] : 0   (column padding)
__global__ void pad_cols(const float* __restrict__ src, float* __restrict__ dstp,
                         int N0, int NP, int total) {
    int i = blockIdx.x * blockDim.x + threadIdx.x;
    if (i >= total) return;
    int r = i / NP, c = i - r * NP;
    dstp[i] = (c < N0) ? src[r * N0 + c] : 0.f;
}

// agg[i, f] = (1+eps)*x[i, f] for f < F, 0 in pad columns (out stride ldout)
__global__ void init_scaled(float* __restrict__ dstp, const float* __restrict__ xin,
                            const float* __restrict__ eps, int F, int ldout, long long n) {
    long long i = (long long)blockIdx.x * blockDim.x + threadIdx.x;  // over N*ldout
    if (i >= n) return;
    int node = (int)(i / ldout);
    int f    = (int)(i - (long long)node * ldout);
    dstp[i] = (f < F) ? (1.0f + *eps) * xin[(long long)node * F + f] : 0.f;
}

// edge-parallel scatter-add: thread = (edge, 4-feature chunk)
__global__ void scatter_add(float* __restrict__ agg, const float* __restrict__ xin,
                            const int* __restrict__ src, const int* __restrict__ dst,
                            int F, int ldin, int ldout, int cpe) {
    long long tid = (long long)blockIdx.x * blockDim.x + threadIdx.x;
    long long total = (long long)N_EDGES * cpe;
    if (tid >= total) return;
    int e = (int)(tid / cpe);
    int c = (int)(tid - (long long)e * cpe);
    int s = src[e], d = dst[e];
    const float* xs = xin + (long long)s * ldin;
    float*       ad = agg + (long long)d * ldout;
    int f0 = c * 4;
#pragma unroll
    for (int j = 0; j < 4; ++j) {
        int f = f0 + j;
        if (f < F) atomicAdd(&ad[f], xs[f]);
    }
}

// ---------------------------------------------------------------- WMMA GEMM
// Out[M x (NT*16)] = act(A[M x K] @ W[K x (NT*16)] + bias), fp32 WMMA 16x16x4.
// All shapes padded so there are NO guards: straight-line loads + wmma.
// A frag (ISA 16x4 f32): lane m = lane%16, elems = K[2*(lane/16) + {0,1}].
// B frag mirrored: lane n = lane%16, same K pair of column n.
// C/D: vgpr v, lane l -> row v + 8*(l/16), col l%16.
template <int K, int LDA, int LDW, int NT, int LDO, bool RELU>
__global__ __launch_bounds__(64) void gemm_wmma(
    const float* __restrict__ A, const float* __restrict__ W,
    const float* __restrict__ bias, float* __restrict__ Out)
{
    const int lane  = threadIdx.x & 31;
    const int wave  = threadIdx.x >> 5;
    const int tile  = blockIdx.x * 2 + wave;
    const int r0    = tile * 16;
    const int mrow  = lane & 15;          // M (A) / N (B,D) within tile
    const int khalf = (lane >> 4) * 2;    // K sub-offset of this lane-half

    v8f acc[NT];
#pragma unroll
    for (int t = 0; t < NT; ++t) acc[t] = (v8f){0.f,0.f,0.f,0.f,0.f,0.f,0.f,0.f};

    const float* Arow = A + (long long)(r0 + mrow) * LDA + khalf;
    const float* Wcol = W + (long long)khalf * LDW + mrow;

#pragma unroll 2
    for (int kc = 0; kc < K / 4; ++kc) {
        const v2f a = *(const v2f*)(Arow + kc * 4);
#pragma unroll
        for (int t = 0; t < NT; ++t) {
            v2f b;
            b.x = Wcol[(long long)kc * 4 * LDW + t * 16];
            b.y = Wcol[(long long)kc * 4 * LDW + LDW + t * 16];
            acc[t] = __builtin_amdgcn_wmma_f32_16x16x4_f32(
                false, a, false, b, (short)0, acc[t], false, false);
        }
    }

    const int rquad = (lane >> 4) * 8;
    float* orow = Out + (long long)(r0 + rquad) * LDO + mrow;
#pragma unroll
    for (int t = 0; t < NT; ++t) {
        const float bv = bias[t * 16 + mrow];
#pragma unroll
        for (int v = 0; v < 8; ++v) {
            float dv = acc[t][v] + bv;
            if (RELU) dv = fmaxf(dv, 0.f);
            orow[(long long)v * LDO + t * 16] = dv;
        }
    }
}

// ---------------------------------------------------------------- BatchNorm
#define BN_ROWS 256
__global__ void bn_stats(const float* __restrict__ h, float* __restrict__ stats, int nrows) {
    const int f  = threadIdx.x;               // 128 features
    const int r0 = blockIdx.x * BN_ROWS;
    int rend = r0 + BN_ROWS; if (rend > nrows) rend = nrows;
    float s = 0.f, s2 = 0.f;
    for (int r = r0; r < rend; ++r) {
        float v = h[(long long)r * HIDDEN + f];
        s += v; s2 += v * v;
    }
    atomicAdd(&stats[f], s);
    atomicAdd(&stats[HIDDEN + f], s2);
}

__global__ void bn_apply(float* __restrict__ h, const float* __restrict__ stats,
                         const float* __restrict__ g, const float* __restrict__ beta,
                         long long n, float inv_n) {
    long long i = (long long)blockIdx.x * blockDim.x + threadIdx.x;
    if (i >= n) return;
    int f = (int)(i & (HIDDEN - 1));
    float mu  = stats[f] * inv_n;
    float var = stats[HIDDEN + f] * inv_n - mu * mu;
    float inv = rsqrtf(var + BN_EPS);
    h[i] = (h[i] - mu) * inv * g[f] + beta[f];
}

// ---------------------------------------------------------------- pooling
__global__ void pool_accum(float* __restrict__ pooled, float* __restrict__ counts,
                           const float* __restrict__ h, const int* __restrict__ batch) {
    long long tid = (long long)blockIdx.x * blockDim.x + threadIdx.x;
    long long total = (long long)N_NODES * 32;
    if (tid >= total) return;
    int node = (int)(tid >> 5);
    int c    = (int)(tid & 31);
    int g    = batch[node];
    const float* hp = h + (long long)node * HIDDEN + c * 4;
    float*       pp = pooled + (long long)g * HIDDEN + c * 4;
#pragma unroll
    for (int j = 0; j < 4; ++j) atomicAdd(&pp[j], hp[j]);
    if (c == 0) atomicAdd(&counts[g], 1.0f);
}

__global__ void pool_div(float* __restrict__ pooled, const float* __restrict__ counts) {
    int i = blockIdx.x * blockDim.x + threadIdx.x;
    if (i >= N_GRAPHS * HIDDEN) return;
    int g = i >> 7;
    pooled[i] /= fmaxf(counts[g], 1.0f);
}

// ---------------------------------------------------------------- log_softmax
__global__ void logsoftmax_rows(const float* __restrict__ z, float* __restrict__ out) {
    int r = blockIdx.x * blockDim.x + threadIdx.x;
    if (r >= N_GRAPHS) return;
    const float* zr = z + (long long)r * OUT_PAD;   // padded stride, first 50 valid
    float m = -3.0e38f;
    for (int c = 0; c < OUT_DIM; ++c) m = fmaxf(m, zr[c]);
    float s = 0.f;
    for (int c = 0; c < OUT_DIM; ++c) s += expf(zr[c] - m);
    float l = logf(s);
    float* o = out + (long long)r * OUT_DIM;
    for (int c = 0; c < OUT_DIM; ++c) o[c] = zr[c] - m - l;
}

// ---------------------------------------------------------------- launcher
static inline unsigned blocks_for(long long n, int bs) {
    return (unsigned)((n + bs - 1) / bs);
}

extern "C" void kernel_launch(void* const* d_in, const int* in_sizes, int n_in,
                              void* d_out, int out_size, void* d_ws, size_t ws_size,
                              hipStream_t stream) {
    const float* x        = (const float*)d_in[0];
    const int*   ei       = (const int*)d_in[1];
    const int*   batch    = (const int*)d_in[2];
    const float* eps0     = (const float*)d_in[3];
    const float* W1_0     = (const float*)d_in[4];
    const float* b1_0     = (const float*)d_in[5];
    const float* W2_0     = (const float*)d_in[6];
    const float* b2_0     = (const float*)d_in[7];
    const float* g0       = (const float*)d_in[8];
    const float* beta0    = (const float*)d_in[9];
    const float* cv_eps   = (const float*)d_in[10];
    const float* cv_W1    = (const float*)d_in[11];
    const float* cv_b1    = (const float*)d_in[12];
    const float* cv_W2    = (const float*)d_in[13];
    const float* cv_b2    = (const float*)d_in[14];
    const float* cv_g     = (const float*)d_in[15];
    const float* cv_beta  = (const float*)d_in[16];
    const float* lin1_W   = (const float*)d_in[17];
    const float* lin1_b   = (const float*)d_in[18];
    const float* lin2_W   = (const float*)d_in[19];
    const float* lin2_b   = (const float*)d_in[20];
    (void)in_sizes; (void)n_in; (void)out_size; (void)ws_size;

    const int* src = ei;
    const int* dst = ei + N_EDGES;

    const size_t NB = (size_t)N_NODES * HIDDEN * sizeof(float); // 51.2 MB
    char* w = (char*)d_ws;
    float* A      = (float*)(w);
    float* B      = (float*)(w + NB);
    float* C      = (float*)(w + 2 * NB);
    float* stats  = (float*)(w + 3 * NB);          // 256 floats: sum | sumsq
    float* pooled = stats + 2 * HIDDEN;            // 1024*128
    float* counts = pooled + N_GRAPHS * HIDDEN;    // 1024
    float* z1     = counts + N_GRAPHS;             // 1024*128
    float* z2     = z1 + N_GRAPHS * HIDDEN;        // 1024*64
    float* w1pad  = z2 + N_GRAPHS * OUT_PAD;       // 80*128
    float* w2pad  = w1pad + IN_PAD * HIDDEN;       // 128*64
    float* b2pad  = w2pad + HIDDEN * OUT_PAD;      // 64
    float* out    = (float*)d_out;

    const int TB = 256;
    const unsigned gemm_grid_nodes = (N_NODES / 16) / 2;  // 3125 blocks x 2 waves
    const unsigned gemm_grid_head  = (N_GRAPHS / 16) / 2; // 32

    // ---------- pad weights once (zero rows/cols: exact for GEMM) ----------
    copy_zero_tail<<<blocks_for(IN_PAD * HIDDEN, TB), TB, 0, stream>>>(
        W1_0, w1pad, IN_DIM * HIDDEN, IN_PAD * HIDDEN);
    pad_cols<<<blocks_for(HIDDEN * OUT_PAD, TB), TB, 0, stream>>>(
        lin2_W, w2pad, OUT_DIM, OUT_PAD, HIDDEN * OUT_PAD);
    copy_zero_tail<<<1, OUT_PAD, 0, stream>>>(lin2_b, b2pad, OUT_DIM, OUT_PAD);

    // ---------- layer 0 (F = 79, padded to 80) ----------
    {
        long long n0 = (long long)N_NODES * IN_PAD;
        init_scaled<<<blocks_for(n0, TB), TB, 0, stream>>>(A, x, eps0, IN_DIM, IN_PAD, n0);
        int cpe = (IN_DIM + 3) / 4; // 20
        scatter_add<<<blocks_for((long long)N_EDGES * cpe, TB), TB, 0, stream>>>(
            A, x, src, dst, IN_DIM, IN_DIM, IN_PAD, cpe);
        gemm_wmma<IN_PAD, IN_PAD, HIDDEN, 8, HIDDEN, true><<<gemm_grid_nodes, 64, 0, stream>>>(
            A, w1pad, b1_0, B);
        gemm_wmma<HIDDEN, HIDDEN, HIDDEN, 8, HIDDEN, true><<<gemm_grid_nodes, 64, 0, stream>>>(
            B, W2_0, b2_0, C);
        zero_f32<<<1, 256, 0, stream>>>(stats, 2 * HIDDEN);
        bn_stats<<<blocks_for(N_NODES, BN_ROWS), HIDDEN, 0, stream>>>(C, stats, N_NODES);
        long long nh = (long long)N_NODES * HIDDEN;
        bn_apply<<<blocks_for(nh, TB), TB, 0, stream>>>(C, stats, g0, beta0, nh, 1.0f / N_NODES);
    }

    // ---------- layers 1,2 (F = 128), buffer rotation C->A->C ----------
    float* cur = C;
    for (int l = 0; l < 2; ++l) {
        float* h0 = (l == 0) ? A : C;   // scatter target; layer output reuses it
        const float* W1 = cv_W1 + (size_t)l * HIDDEN * HIDDEN;
        const float* b1 = cv_b1 + (size_t)l * HIDDEN;
        const float* W2 = cv_W2 + (size_t)l * HIDDEN * HIDDEN;
        const float* b2 = cv_b2 + (size_t)l * HIDDEN;
        const float* gg = cv_g    + (size_t)l * HIDDEN;
        const float* bb = cv_beta + (size_t)l * HIDDEN;

        long long nh = (long long)N_NODES * HIDDEN;
        init_scaled<<<blocks_for(nh, TB), TB, 0, stream>>>(h0, cur, cv_eps + l, HIDDEN, HIDDEN, nh);
        scatter_add<<<blocks_for((long long)N_EDGES * 32, TB), TB, 0, stream>>>(
            h0, cur, src, dst, HIDDEN, HIDDEN, HIDDEN, 32);
        gemm_wmma<HIDDEN, HIDDEN, HIDDEN, 8, HIDDEN, true><<<gemm_grid_nodes, 64, 0, stream>>>(
            h0, W1, b1, B);
        gemm_wmma<HIDDEN, HIDDEN, HIDDEN, 8, HIDDEN, true><<<gemm_grid_nodes, 64, 0, stream>>>(
            B, W2, b2, h0);
        zero_f32<<<1, 256, 0, stream>>>(stats, 2 * HIDDEN);
        bn_stats<<<blocks_for(N_NODES, BN_ROWS), HIDDEN, 0, stream>>>(h0, stats, N_NODES);
        bn_apply<<<blocks_for(nh, TB), TB, 0, stream>>>(h0, stats, gg, bb, nh, 1.0f / N_NODES);
        cur = h0;
    }

    // ---------- global mean pool ----------
    zero_f32<<<blocks_for(N_GRAPHS * HIDDEN + N_GRAPHS, TB), TB, 0, stream>>>(
        pooled, N_GRAPHS * HIDDEN + N_GRAPHS);
    pool_accum<<<blocks_for((long long)N_NODES * 32, TB), TB, 0, stream>>>(
        pooled, counts, cur, batch);
    pool_div<<<blocks_for(N_GRAPHS * HIDDEN, TB), TB, 0, stream>>>(pooled, counts);

    // ---------- classifier head ----------
    gemm_wmma<HIDDEN, HIDDEN, HIDDEN, 8, HIDDEN, true><<<gemm_grid_head, 64, 0, stream>>>(
        pooled, lin1_W, lin1_b, z1);
    gemm_wmma<HIDDEN, HIDDEN, OUT_PAD, 4, OUT_PAD, false><<<gemm_grid_head, 64, 0, stream>>>(
        z1, w2pad, b2pad, z2);
    logsoftmax_rows<<<blocks_for(N_GRAPHS, TB), TB, 0, stream>>>(z2, out);
}